// CompactLoss_13864154431845
// MI455X (gfx1250) — compile-verified
//
#include <hip/hip_runtime.h>
#include <math.h>

// Problem constants (from reference)
#define G 16
#define D 512
#define NB 32768            // BATCH
#define TILES_PER_G (NB/16) // 2048 tiles of 16 samples
#define NBLOCKS 8192        // 8192 blocks * 4 waves = 32768 wave-tasks = G * TILES_PER_G

// Workspace layout (in floats)
#define WS_CNORM 0            // [0, 8192): normalized centers (16 x 512)
#define WS_CSQ   (G * D)      // [8192, 8208): ||c_norm_g||^2 per group
#define WS_PART  (G * D + 16) // [8208, 8208+8192): per-block partial sums

typedef float v2f __attribute__((ext_vector_type(2)));
typedef float v8f __attribute__((ext_vector_type(8)));

// ---------------------------------------------------------------------------
// Kernel A: L2-normalize centers (eps = 1e-12), store c_norm and ||c_norm||^2.
// 16 blocks (one per group) x 512 threads (one per feature).
// ---------------------------------------------------------------------------
__global__ void prep_centers(const float* __restrict__ centers, float* __restrict__ ws) {
    __shared__ float red[D];
    const int g = blockIdx.x;
    const int d = threadIdx.x;
    const float v = centers[g * D + d];
    red[d] = v * v;
    __syncthreads();
    for (int s = D / 2; s > 0; s >>= 1) {
        if (d < s) red[d] += red[d + s];
        __syncthreads();
    }
    const float sumsq = red[0];
    const float denom = fmaxf(sqrtf(sumsq), 1e-12f);
    ws[WS_CNORM + g * D + d] = v / denom;
    if (d == 0) ws[WS_CSQ + g] = sumsq / (denom * denom);
}

// ---------------------------------------------------------------------------
// Kernel B: streaming main kernel. One wave owns (group g, 16 samples).
//   A (16x4 f32)  = gf[g, b0:b0+16, k0:k0+4]   (per-lane float2, non-temporal)
//   B (4x16 f32)  = c_norm_g[k0:k0+4] broadcast into all 16 columns (from LDS)
//   acc (16x16)   accumulates cross[b] in every column via v_wmma_f32_16x16x4_f32
// ||gf||^2 accumulated in VALU on the same registers.
// ---------------------------------------------------------------------------
__global__ void __launch_bounds__(128)
main_kernel(const float* __restrict__ gf, const float* __restrict__ ws,
            float* __restrict__ partials) {
    __shared__ float cnorm_s[D];
    __shared__ float red[128];

    const int tid  = threadIdx.x;
    const int wid  = tid >> 5;        // wave in block (0..3)
    const int lane = tid & 31;
    const int row  = lane & 15;       // M index this lane carries in the A frag
    const int kq   = lane >> 4;       // 0: K={0,1}, 1: K={2,3}

    // Task mapping: consecutive waves -> consecutive sample tiles, same group
    // (TILES_PER_G % 4 == 0, so all 4 waves of a block share g).
    const int W  = blockIdx.x * 4 + wid;
    const int g  = W >> 11;           // / TILES_PER_G
    const int t  = W & (TILES_PER_G - 1);
    const int b0 = t * 16;

    // Stage this group's normalized center into LDS once per block.
    for (int i = tid; i < D; i += 128) cnorm_s[i] = ws[WS_CNORM + g * D + i];
    __syncthreads();

    const float* Arow = gf + ((size_t)g * NB + (size_t)(b0 + row)) * D + 2 * kq;
    const float* Brow = cnorm_s + 2 * kq;

    v8f acc = {};     // C/D: 16x16 f32, all columns equal = cross[b]
    float sq = 0.0f;  // per-lane partial of sum(gf^2) for this lane's row

    #pragma unroll 4
    for (int k0 = 0; k0 < D; k0 += 4) {
        // A fragment: lane L -> A[row, k0 + 2*kq + {0,1}] ; stream past L2 (NT)
        v2f a = __builtin_nontemporal_load((const v2f*)(Arow + k0));
        // B fragment: broadcast columns -> lane needs c[k0 + 2*kq + {0,1}] (LDS)
        v2f b = *(const v2f*)(Brow + k0);
        sq += a.x * a.x + a.y * a.y;
        // D = A x B + C, full fp32 precision
        acc = __builtin_amdgcn_wmma_f32_16x16x4_f32(
            /*neg_a=*/false, a, /*neg_b=*/false, b,
            /*c_mod=*/(short)0, acc, /*reuse_a=*/false, /*reuse_b=*/false);
    }

    // ||gf[row]||^2 : lane m holds K%4<2 half, lane m+16 the other half.
    const float gsq = sq + __shfl_xor(sq, 16, 32);
    const float csq = ws[WS_CSQ + g];  // uniform per wave (L2-resident)

    // C layout: VGPR r -> rows r (lanes 0-15) and r+8 (lanes 16-31); columns
    // are identical, so lane holds cross[r + 8*kq] in acc[r].
    float s = 0.0f;
    #pragma unroll
    for (int r = 0; r < 8; ++r) {
        const int   m     = r + 8 * kq;
        const float gs    = __shfl(gsq, m, 32);     // g_sq for row m
        const float cross = acc[r];
        float dmm = gs + csq - 2.0f * cross;
        dmm = fminf(fmaxf(dmm, 1e-12f), 1e12f);     // clip(1e-12, 1e12)
        s += dmm;
    }
    // Lanes within a half are redundant copies; count rows 0-7 once (lane 0)
    // and rows 8-15 once (lane 16).
    const float contrib = (row == 0) ? s : 0.0f;

    red[tid] = contrib;
    __syncthreads();
    for (int st = 64; st > 0; st >>= 1) {
        if (tid < st) red[tid] += red[tid + st];
        __syncthreads();
    }
    if (tid == 0) partials[blockIdx.x] = red[0];
}

// ---------------------------------------------------------------------------
// Kernel C: deterministic fixed-order reduction of block partials + scaling.
// ---------------------------------------------------------------------------
__global__ void finalize(const float* __restrict__ partials, float* __restrict__ out) {
    __shared__ float red[256];
    const int tid = threadIdx.x;
    float s = 0.0f;
    for (int i = tid; i < NBLOCKS; i += 256) s += partials[i];
    red[tid] = s;
    __syncthreads();
    for (int st = 128; st > 0; st >>= 1) {
        if (tid < st) red[tid] += red[tid + st];
        __syncthreads();
    }
    if (tid == 0) out[0] = red[0] * (1.0f / ((float)G * (float)NB));
}

extern "C" void kernel_launch(void* const* d_in, const int* in_sizes, int n_in,
                              void* d_out, int out_size, void* d_ws, size_t ws_size,
                              hipStream_t stream) {
    const float* gf      = (const float*)d_in[0];  // (16, 32768, 512) f32
    const float* centers = (const float*)d_in[1];  // (16, 512) f32
    float* ws  = (float*)d_ws;
    float* out = (float*)d_out;

    prep_centers<<<G, D, 0, stream>>>(centers, ws);
    main_kernel<<<NBLOCKS, 128, 0, stream>>>(gf, ws, ws + WS_PART);
    finalize<<<1, 256, 0, stream>>>(ws + WS_PART, out);
}